// Model4_9620726743220
// MI455X (gfx1250) — compile-verified
//
#include <hip/hip_runtime.h>
#include <math.h>

typedef __attribute__((ext_vector_type(2))) float v2f;
typedef __attribute__((ext_vector_type(8))) float v8f;

// a/g/acc: [Npad,16]; bm: [Npad,19]; concat K = 34 padded -> 36.

__global__ void k_zero(float* acc, unsigned* degU, float* partial, int npad16, int N) {
    int i = blockIdx.x * blockDim.x + threadIdx.x;
    if (i < npad16) acc[i] = 0.0f;
    if (i < N) degU[i] = 0u;
    if (i == 0) partial[0] = 0.0f;
}

// Zero-padded B tile: WgPad[k*16+n] = (k<34 && n<15) ? Wg[n][k] : 0
__global__ void k_padWg(const float* Wg, float* WgPad) {
    int i = blockIdx.x * blockDim.x + threadIdx.x;
    if (i < 36 * 16) {
        int k = i >> 4, n = i & 15;
        WgPad[i] = (k < 34 && n < 15) ? Wg[n * 34 + k] : 0.0f;
    }
}

__global__ void k_deg(const long long* col, unsigned* degU, int E) {
    int e = blockIdx.x * blockDim.x + threadIdx.x;
    if (e < E) atomicAdd(&degU[(int)col[e]], 1u);
}

// dinv, b = cat(x1, tile(x2)) @ W4.T + b4, a = relu(x1 @ W1.T + b1)
__global__ void k_prep(const float* x1, const float* x2, const float* W1, const float* b1,
                       const float* W4, const float* b4, const unsigned* degU,
                       float* a, float* bm, float* dinv, int N, int NB) {
    __shared__ float sW1[225], sb1[15], sW4[361], sb4[19];
    for (int t = threadIdx.x; t < 225; t += blockDim.x) sW1[t] = W1[t];
    for (int t = threadIdx.x; t < 15; t += blockDim.x) sb1[t] = b1[t];
    for (int t = threadIdx.x; t < 361; t += blockDim.x) sW4[t] = W4[t];
    for (int t = threadIdx.x; t < 19; t += blockDim.x) sb4[t] = b4[t];
    __syncthreads();
    int i = blockIdx.x * blockDim.x + threadIdx.x;
    if (i >= N) return;
    dinv[i] = rsqrtf((float)(degU[i] + 1u));  // self-loop => deg >= 1
    float xv[19];
#pragma unroll
    for (int j = 0; j < 15; ++j) xv[j] = x1[i * 15 + j];
    int ib = i % NB;  // tile(x2, (20,1))
#pragma unroll
    for (int j = 0; j < 4; ++j) xv[15 + j] = x2[ib * 4 + j];
#pragma unroll
    for (int o = 0; o < 19; ++o) {
        float s = sb4[o];
#pragma unroll
        for (int j = 0; j < 19; ++j) s += xv[j] * sW4[o * 19 + j];
        bm[i * 19 + o] = s;
    }
#pragma unroll
    for (int o = 0; o < 15; ++o) {
        float s = sb1[o];
#pragma unroll
        for (int j = 0; j < 15; ++j) s += xv[j] * sW1[o * 15 + j];
        a[i * 16 + o] = fmaxf(s, 0.0f);
    }
    a[i * 16 + 15] = 0.0f;
}

// g = dinv * (cat(a,b) @ Wg.T). 256 threads = 8 waves = 128 nodes per block.
// All arrays padded to Npad rows => completely branch-free (padding rows carry
// garbage that is never consumed). Inner loop: ds loads + V_WMMA_F32_16X16X4_F32.
__global__ void __launch_bounds__(256) k_hg_wmma(const float* a, const float* bm,
                                                 const float* dinv, const float* WgPad,
                                                 float* g) {
    __shared__ float sF[128 * 36];  // [node_local][k]
    __shared__ float sB[36 * 16];   // [k][n], pre-padded

    int blockBase = blockIdx.x * 128;
    const float* a_blk = a + (size_t)blockBase * 16;
    const float* bm_blk = bm + (size_t)blockBase * 19;

    // B tile: plain copy of pre-padded weights (576 elems)
    for (int i = threadIdx.x; i < 36 * 16; i += 256) sB[i] = WgPad[i];
    // a-part: cols 0..14 (128*15 = 1920 elems)
    for (int i = threadIdx.x; i < 128 * 15; i += 256) {
        int nl = i / 15;
        int k = i - nl * 15;
        sF[nl * 36 + k] = a_blk[nl * 16 + k];
    }
    // bm-part: cols 15..33 (128*19 = 2432 elems; global side is fully linear)
    for (int i = threadIdx.x; i < 128 * 19; i += 256) {
        int nl = i / 19;
        int k = i - nl * 19;
        sF[nl * 36 + 15 + k] = bm_blk[i];
    }
    // pad cols 34,35 (256 elems: exactly one per thread)
    {
        int nl = threadIdx.x >> 1;
        sF[nl * 36 + 34 + (threadIdx.x & 1)] = 0.0f;
    }
    __syncthreads();

    int lane = threadIdx.x & 31;
    int mlane = lane & 15;
    int khalf = (lane >> 4) << 1;            // lanes 16-31: K+2, K+3 (32-bit A layout)
    int wlocal = (threadIdx.x >> 5) << 4;    // wave's 16-node group within block
    int arow = (wlocal + mlane) * 36;

    v8f acc = {};
#pragma unroll
    for (int t = 0; t < 9; ++t) {
        int k0 = 4 * t + khalf;
        v2f A, B;
        A.x = sF[arow + k0];
        A.y = sF[arow + k0 + 1];
        B.x = sB[k0 * 16 + mlane];
        B.y = sB[(k0 + 1) * 16 + mlane];
        acc = __builtin_amdgcn_wmma_f32_16x16x4_f32(
            false, A, false, B, (short)0, acc, false, false);
    }

    int moff = (lane >> 4) << 3;  // C/D: lanes 16-31 hold M+8
    int base = blockBase + wlocal + moff;
    float4 d0 = *(const float4*)&dinv[base];      // 32B-aligned (base % 8 == 0)
    float4 d1 = *(const float4*)&dinv[base + 4];
    float dv[8] = {d0.x, d0.y, d0.z, d0.w, d1.x, d1.y, d1.z, d1.w};
#pragma unroll
    for (int i = 0; i < 8; ++i) {
        g[(size_t)(base + i) * 16 + mlane] = dv[i] * acc[i];  // col 15 is 0
    }
}

// acc[col] += g[row] for all edges (dinv[row] already folded into g)
__global__ void k_scatter(const long long* row, const long long* col,
                          const float* g, float* acc, int E) {
    int e = blockIdx.x * blockDim.x + threadIdx.x;
    if (e >= E) return;
    int r = (int)row[e];
    int c = (int)col[e];
    const float4* gr = (const float4*)(g + (size_t)r * 16);  // 64B-aligned rows
    float4 v0 = gr[0], v1 = gr[1], v2 = gr[2], v3 = gr[3];
    float* ac = acc + (size_t)c * 16;
    atomicAdd(ac + 0,  v0.x); atomicAdd(ac + 1,  v0.y);
    atomicAdd(ac + 2,  v0.z); atomicAdd(ac + 3,  v0.w);
    atomicAdd(ac + 4,  v1.x); atomicAdd(ac + 5,  v1.y);
    atomicAdd(ac + 6,  v1.z); atomicAdd(ac + 7,  v1.w);
    atomicAdd(ac + 8,  v2.x); atomicAdd(ac + 9,  v2.y);
    atomicAdd(ac + 10, v2.z); atomicAdd(ac + 11, v2.w);
    atomicAdd(ac + 12, v3.x); atomicAdd(ac + 13, v3.y);
    atomicAdd(ac + 14, v3.z);
}

// a = relu(dinv[c] * (acc[c] + g[c]) + bg); re-zero acc for next iteration
__global__ void k_update(const float* g, float* acc, const float* dinv,
                         const float* bg, float* a, int N) {
    int idx = blockIdx.x * blockDim.x + threadIdx.x;
    if (idx >= N * 16) return;
    int i = idx >> 4, k = idx & 15;
    if (k < 15) {
        float v = dinv[i] * (acc[idx] + g[idx]) + bg[k];
        a[idx] = fmaxf(v, 0.0f);
        acc[idx] = 0.0f;
    } else {
        a[idx] = 0.0f;
    }
}

// per-node cat(a,b) @ W3.T + b3 ; block-reduce; atomic into partial sum
__global__ void k_final(const float* a, const float* bm, const float* W3, const float* b3,
                        float* partial, int N) {
    __shared__ float sW[34];
    __shared__ float red[256];
    for (int t = threadIdx.x; t < 34; t += blockDim.x) sW[t] = W3[t];
    __syncthreads();
    int i = blockIdx.x * blockDim.x + threadIdx.x;
    float s = 0.0f;
    if (i < N) {
        s = b3[0];
#pragma unroll
        for (int k = 0; k < 15; ++k) s += a[i * 16 + k] * sW[k];
#pragma unroll
        for (int k = 0; k < 19; ++k) s += bm[i * 19 + k] * sW[15 + k];
    }
    red[threadIdx.x] = s;
    __syncthreads();
    for (int off = blockDim.x / 2; off > 0; off >>= 1) {
        if ((int)threadIdx.x < off) red[threadIdx.x] += red[threadIdx.x + off];
        __syncthreads();
    }
    if (threadIdx.x == 0) atomicAdd(partial, red[0]);
}

__global__ void k_finish(const float* partial, float* out, int N) {
    if (threadIdx.x == 0 && blockIdx.x == 0) out[0] = tanhf(partial[0] / (float)N);
}

extern "C" void kernel_launch(void* const* d_in, const int* in_sizes, int n_in,
                              void* d_out, int out_size, void* d_ws, size_t ws_size,
                              hipStream_t stream) {
    const float* x1 = (const float*)d_in[0];
    const float* x2 = (const float*)d_in[1];
    const long long* edges = (const long long*)d_in[2];
    const float* W1 = (const float*)d_in[3];
    const float* b1 = (const float*)d_in[4];
    const float* Wg = (const float*)d_in[5];
    const float* bg = (const float*)d_in[6];
    const float* W3 = (const float*)d_in[7];
    const float* b3 = (const float*)d_in[8];
    const float* W4 = (const float*)d_in[9];
    const float* b4 = (const float*)d_in[10];
    float* out = (float*)d_out;

    int N = in_sizes[0] / 15;
    int NB = in_sizes[1] / 4;
    int E = in_sizes[2] / 2;
    int Npad = (N + 127) & ~127;  // whole 128-node blocks, branch-free WMMA kernel
    const long long* row = edges;
    const long long* col = edges + E;

    char* ws = (char*)d_ws;
    size_t off = 0;
    auto wsalloc = [&](size_t bytes) -> void* {
        void* p = (void*)(ws + off);
        off += (bytes + 255) & ~(size_t)255;
        return p;
    };
    float* a       = (float*)wsalloc((size_t)Npad * 16 * sizeof(float));
    float* bm      = (float*)wsalloc((size_t)Npad * 19 * sizeof(float));
    float* g       = (float*)wsalloc((size_t)Npad * 16 * sizeof(float));
    float* acc     = (float*)wsalloc((size_t)Npad * 16 * sizeof(float));
    float* dinv    = (float*)wsalloc((size_t)Npad * sizeof(float));
    unsigned* degU = (unsigned*)wsalloc((size_t)N * sizeof(unsigned));
    float* partial = (float*)wsalloc(sizeof(float));
    float* WgPad   = (float*)wsalloc(36 * 16 * sizeof(float));
    (void)ws_size; (void)n_in; (void)out_size;

    const int B = 256;
    int npad16 = Npad * 16;
    int n16 = N * 16;
    k_zero<<<(npad16 + B - 1) / B, B, 0, stream>>>(acc, degU, partial, npad16, N);
    k_padWg<<<3, B, 0, stream>>>(Wg, WgPad);
    k_deg<<<(E + B - 1) / B, B, 0, stream>>>(col, degU, E);
    k_prep<<<(N + B - 1) / B, B, 0, stream>>>(x1, x2, W1, b1, W4, b4, degU, a, bm, dinv, N, NB);

    int hgBlocks = Npad / 128;
    for (int it = 0; it < 5; ++it) {
        k_hg_wmma<<<hgBlocks, B, 0, stream>>>(a, bm, dinv, WgPad, g);
        k_scatter<<<(E + B - 1) / B, B, 0, stream>>>(row, col, g, acc, E);
        k_update<<<(n16 + B - 1) / B, B, 0, stream>>>(g, acc, dinv, bg, a, N);
    }

    k_final<<<(N + B - 1) / B, B, 0, stream>>>(a, bm, W3, b3, partial, N);
    k_finish<<<1, 32, 0, stream>>>(partial, out, N);
}